// FractalMoE_71717363908753
// MI455X (gfx1250) — compile-verified
//
#include <hip/hip_runtime.h>

#define NEXP   32
#define H      512
#define H2     1024
#define NTOK   2048
#define MTILE  32
#define XPAD   4
#define HPAD   4

typedef __attribute__((ext_vector_type(2))) float v2f;
typedef __attribute__((ext_vector_type(8))) float v8f;
typedef __attribute__((ext_vector_type(4))) int   v4i;

#if defined(__has_builtin)
#if __has_builtin(__builtin_amdgcn_global_load_async_to_lds_b128)
#define HAVE_ASYNC_LDS 1
#endif
#if __has_builtin(__builtin_amdgcn_s_wait_asynccnt)
#define HAVE_WAIT_ASYNC 1
#endif
#endif
#ifndef HAVE_ASYNC_LDS
#define HAVE_ASYNC_LDS 0
#endif
#ifndef HAVE_WAIT_ASYNC
#define HAVE_WAIT_ASYNC 0
#endif

// ---------------------------------------------------------------------------
// Kernel 0: zero the per-expert token counters (re-run every launch).
// ---------------------------------------------------------------------------
__global__ void moe_init_kernel(int* __restrict__ counts) {
    if (threadIdx.x < NEXP) counts[threadIdx.x] = 0;
}

// ---------------------------------------------------------------------------
// Kernel 1: gating. One wave32 per token, lane == expert.
// ---------------------------------------------------------------------------
__global__ __launch_bounds__(256)
void moe_gate_kernel(const float* __restrict__ x,
                     const float* __restrict__ gate_w,
                     int*   __restrict__ counts,
                     int*   __restrict__ routing,   // [NEXP][NTOK] packed (tok<<2)|rank
                     float* __restrict__ tokw)      // [NEXP][NTOK]
{
    const int lane = threadIdx.x & 31;       // expert id
    const int wv   = threadIdx.x >> 5;
    const int n    = blockIdx.x * 8 + wv;    // token id (grid = NTOK/8)

    const float* xr = x      + (size_t)n    * H;
    const float* gr = gate_w + (size_t)lane * H;

    float acc = 0.f;
#pragma unroll 4
    for (int h = 0; h < H; h += 4) {
        float4 xv = *(const float4*)(xr + h);
        float4 gv = *(const float4*)(gr + h);
        acc += xv.x * gv.x + xv.y * gv.y + xv.z * gv.z + xv.w * gv.w;
    }

    // softmax over the 32 lanes (wave32 shuffles)
    float m = acc;
    for (int off = 16; off; off >>= 1) m = fmaxf(m, __shfl_xor(m, off, 32));
    float ex = __expf(acc - m);
    float s = ex;
    for (int off = 16; off; off >>= 1) s += __shfl_xor(s, off, 32);
    float p = ex / s;                         // strictly > 0

    // top-4 via 4 rounds of 64-bit key max-reduce; excluded lanes use key 0
    unsigned int kv = __float_as_uint(p);
    int   sele[4];
    float selp[4];
#pragma unroll
    for (int k = 0; k < 4; ++k) {
        unsigned long long key =
            ((unsigned long long)kv << 6) | (unsigned long long)lane;
        for (int off = 16; off; off >>= 1) {
            unsigned long long o = __shfl_xor(key, off, 32);
            key = (o > key) ? o : key;
        }
        int ek = (int)(key & 63ull);
        sele[k] = ek;
        selp[k] = __uint_as_float((unsigned int)(key >> 6));
        if (lane == ek) kv = 0u;
    }
    float wsum = selp[0] + selp[1] + selp[2] + selp[3];

    if (lane < 4) {
        int   ek = sele[lane];
        float w  = selp[lane] / wsum;
        int pos = atomicAdd(&counts[ek], 1);
        routing[ek * NTOK + pos] = (n << 2) | lane;
        tokw   [ek * NTOK + pos] = w;
    }
}

// ---------------------------------------------------------------------------
// Kernel 2: routed expert FFN, fp32 WMMA 16x16x4, M-tile = 32 tokens.
// Block = 512 threads (16 waves). Two 16-row sub-tiles share every B load.
// ---------------------------------------------------------------------------
__global__ __launch_bounds__(512)
void moe_expert_kernel(const float* __restrict__ x,
                       const float* __restrict__ w_gate,
                       const float* __restrict__ w_up,
                       const float* __restrict__ w_down,
                       const int*   __restrict__ counts,
                       const int*   __restrict__ routing,
                       const float* __restrict__ tokw,
                       float* __restrict__ contrib)   // [NTOK][4][H]
{
    extern __shared__ float smem[];
    float* Xs   = smem;                            // MTILE x (H  + XPAD)
    float* Hs   = Xs + MTILE * (H + XPAD);         // MTILE x (H2 + HPAD)
    float* wts  = Hs + MTILE * (H2 + HPAD);        // MTILE
    int*   toks = (int*)(wts + MTILE);             // MTILE
    int*   rnks = toks + MTILE;                    // MTILE

    const int e  = blockIdx.y;
    const int r0 = blockIdx.x * MTILE;
    const int ne = counts[e];
    if (r0 >= ne) return;                          // uniform across block
    const int nrows = (ne - r0 < MTILE) ? (ne - r0) : MTILE;

    const int tid  = threadIdx.x;
    const int lane = tid & 31;
    const int wv   = tid >> 5;                     // 0..15
    const int l15  = lane & 15;
    const int half = lane >> 4;                    // selects K pair

    if (tid < MTILE) {
        int tk = 0, rk = 0; float w = 0.f;
        if (tid < nrows) {
            int packed = routing[e * NTOK + r0 + tid];
            tk = packed >> 2; rk = packed & 3;
            w  = tokw[e * NTOK + r0 + tid];
        }
        toks[tid] = tk; rnks[tid] = rk; wts[tid] = w;
    }
    __syncthreads();

    // ---- gather 32 token rows into LDS (async global->LDS if available) ----
    for (int i = tid; i < MTILE * (H / 4); i += 512) {
        const int row = i >> 7;                    // /(H/4)
        const int col = (i & 127) * 4;
        const float* g = x + (size_t)toks[row] * H + col;   // pad rows -> tok 0
        float* l = &Xs[row * (H + XPAD) + col];
#if HAVE_ASYNC_LDS
        __builtin_amdgcn_global_load_async_to_lds_b128(
            (v4i*)g,
            (__attribute__((address_space(3))) v4i*)(unsigned int)(size_t)l,
            0, 0);
#else
        float4 v = make_float4(0.f, 0.f, 0.f, 0.f);
        if (row < nrows) v = *(const float4*)g;
        *(float4*)l = v;
#endif
    }
#if HAVE_ASYNC_LDS
#if HAVE_WAIT_ASYNC
    __builtin_amdgcn_s_wait_asynccnt(0);
#else
    asm volatile("s_wait_asynccnt 0" ::: "memory");
#endif
#endif
    __syncthreads();

    const float* wg_e  = w_gate + (size_t)e * H2 * H;
    const float* wu_e  = w_up   + (size_t)e * H2 * H;
    const float* xrow0 = &Xs[ l15       * (H + XPAD)];  // rows 0..15
    const float* xrow1 = &Xs[(l15 + 16) * (H + XPAD)];  // rows 16..31

    // ---- Phase 1: gate & up projections + SiLU; 4 N-tiles per wave ----
#pragma unroll
    for (int t = 0; t < 4; ++t) {
        const int col = (wv * 4 + t) * 16 + l15;
        const float* wgp = wg_e + (size_t)col * H;
        const float* wup = wu_e + (size_t)col * H;
        v8f cg0 = {}, cu0 = {}, cg1 = {}, cu1 = {};
#pragma unroll 2
        for (int k0 = 0; k0 < H; k0 += 4) {
            const int kk = k0 + 2 * half;
            v2f a0 = *(const v2f*)(xrow0 + kk);
            v2f a1 = *(const v2f*)(xrow1 + kk);
            v2f bg = *(const v2f*)(wgp + kk);
            v2f bu = *(const v2f*)(wup + kk);
            cg0 = __builtin_amdgcn_wmma_f32_16x16x4_f32(false, a0, false, bg, (short)0, cg0, false, false);
            cu0 = __builtin_amdgcn_wmma_f32_16x16x4_f32(false, a0, false, bu, (short)0, cu0, false, false);
            cg1 = __builtin_amdgcn_wmma_f32_16x16x4_f32(false, a1, false, bg, (short)0, cg1, false, false);
            cu1 = __builtin_amdgcn_wmma_f32_16x16x4_f32(false, a1, false, bu, (short)0, cu1, false, false);
        }
#pragma unroll
        for (int v = 0; v < 8; ++v) {
            const int row = v + 8 * half;
            float g0 = cg0[v], u0 = cu0[v];
            float g1 = cg1[v], u1 = cu1[v];
            // silu(g)*u with fast reciprocal (v_rcp_f32)
            float h0 = g0 * __builtin_amdgcn_rcpf(1.f + __expf(-g0)) * u0;
            float h1 = g1 * __builtin_amdgcn_rcpf(1.f + __expf(-g1)) * u1;
            Hs[ row       * (H2 + HPAD) + col] = h0;
            Hs[(row + 16) * (H2 + HPAD) + col] = h1;
        }
    }
    __syncthreads();

    // ---- Phase 2: down projection; 2 N-tiles per wave ----
    const float* wd_e  = w_down + (size_t)e * H * H2;
    const float* hrow0 = &Hs[ l15       * (H2 + HPAD)];
    const float* hrow1 = &Hs[(l15 + 16) * (H2 + HPAD)];
#pragma unroll
    for (int t = 0; t < 2; ++t) {
        const int col = (wv * 2 + t) * 16 + l15;
        const float* wdp = wd_e + (size_t)col * H2;
        v8f c0 = {}, c1 = {};
#pragma unroll 4
        for (int k0 = 0; k0 < H2; k0 += 4) {
            const int kk = k0 + 2 * half;
            v2f a0 = *(const v2f*)(hrow0 + kk);
            v2f a1 = *(const v2f*)(hrow1 + kk);
            v2f b  = *(const v2f*)(wdp + kk);
            c0 = __builtin_amdgcn_wmma_f32_16x16x4_f32(false, a0, false, b, (short)0, c0, false, false);
            c1 = __builtin_amdgcn_wmma_f32_16x16x4_f32(false, a1, false, b, (short)0, c1, false, false);
        }
#pragma unroll
        for (int v = 0; v < 8; ++v) {
            const int row0 = v + 8 * half;
            const int row1 = row0 + 16;
            if (row0 < nrows) {
                contrib[((size_t)toks[row0] * 4 + rnks[row0]) * H + col] =
                    c0[v] * wts[row0];
            }
            if (row1 < nrows) {
                contrib[((size_t)toks[row1] * 4 + rnks[row1]) * H + col] =
                    c1[v] * wts[row1];
            }
        }
    }
}

// ---------------------------------------------------------------------------
// Kernel 3: combine the 4 ranked contributions per token (fixed order).
// ---------------------------------------------------------------------------
__global__ __launch_bounds__(256)
void moe_combine_kernel(const float* __restrict__ contrib,
                        float* __restrict__ out)
{
    const int i = blockIdx.x * 256 + threadIdx.x;   // over NTOK*H
    const int n = i >> 9;                            // /H
    const int h = i & (H - 1);
    const float* c = contrib + (size_t)n * 4 * H + h;
    out[i] = ((c[0] + c[H]) + c[2 * H]) + c[3 * H];
}

// ---------------------------------------------------------------------------
extern "C" void kernel_launch(void* const* d_in, const int* in_sizes, int n_in,
                              void* d_out, int out_size, void* d_ws, size_t ws_size,
                              hipStream_t stream) {
    const float* x      = (const float*)d_in[0];
    const float* gate_w = (const float*)d_in[1];
    const float* w_gate = (const float*)d_in[2];
    const float* w_up   = (const float*)d_in[3];
    const float* w_down = (const float*)d_in[4];
    float* out = (float*)d_out;

    char* ws = (char*)d_ws;
    float* contrib = (float*)ws;                                   // NTOK*4*H
    size_t off = (size_t)NTOK * 4 * H * sizeof(float);
    int* counts = (int*)(ws + off);  off += 128;                   // 32 ints (padded)
    int* routing = (int*)(ws + off); off += (size_t)NEXP * NTOK * sizeof(int);
    float* tokw = (float*)(ws + off);

    const size_t smem_bytes =
        (MTILE * (H + XPAD) + MTILE * (H2 + HPAD) + MTILE) * sizeof(float) +
        2 * MTILE * sizeof(int);

    moe_init_kernel<<<1, 32, 0, stream>>>(counts);
    moe_gate_kernel<<<NTOK / 8, 256, 0, stream>>>(x, gate_w, counts, routing, tokw);
    moe_expert_kernel<<<dim3(NTOK / MTILE, NEXP), 512, smem_bytes, stream>>>(
        x, w_gate, w_up, w_down, counts, routing, tokw, contrib);
    moe_combine_kernel<<<(NTOK * H) / 256, 256, 0, stream>>>(contrib, out);
}